// MagnusRLC_10900626998019
// MI455X (gfx1250) — compile-verified
//
#include <hip/hip_runtime.h>
#include <math.h>

// Closed-form damped-oscillator trajectory:
//   out[i] = expm(A*(t[i]-t[0])) @ x0,  A = [[0,1],[-w2,-g]]
// expm(A t) = e^{a t} ( c(t) I + sov(t) (A - a I) ),  a = -g/2
//   underdamped : c = cos(mu t),  sov = sin(mu t)/mu,  mu = sqrt(w2 - g^2/4)
//   overdamped  : c = cosh(nu t), sov = sinh(nu t)/nu (via exps, no inf*0)
//   critical    : c = 1,          sov = t
// out0 = u*x0 + v*(g/2*x0 + x1); out1 = u*x1 + v*(-w2*x0 - g/2*x1)
// with u = e*c, v = e*sov. Constant 2x2 folded into w00..w11.
//
// Memory-bound: 16 MB read (t) + 32 MB write (out) -> ~2.1 us floor @ 23.3 TB/s.
// WMMA deliberately absent: the matrix op is a 2x2 matvec (4 FMAs/point);
// a 16x16x4 f32 WMMA mapping is ~6% utilization plus LDS restaging on both
// sides — a strict pessimization on a streaming-store-bound kernel.

__global__ __launch_bounds__(256)
void magnus_rlc_traj_kernel(const float* __restrict__ t,
                            const float* __restrict__ x0v,
                            const float* __restrict__ om2p,
                            const float* __restrict__ gp,
                            float* __restrict__ out,
                            int n)
{
    const float w2 = *om2p;
    const float g  = *gp;
    const float xa = x0v[0];
    const float xb = x0v[1];
    const float t0 = t[0];

    const float alpha = -0.5f * g;
    const float disc  = w2 - 0.25f * g * g;

    // Folded constant matrix W (out = u*[w00,w01] + v*[w10,w11])
    const float w00 = xa;
    const float w01 = xb;
    const float w10 = 0.5f * g * xa + xb;
    const float w11 = -(w2 * xa) - 0.5f * g * xb;

    const int   mode  = (disc > 0.0f) ? 0 : ((disc < 0.0f) ? 1 : 2);
    const float mu    = sqrtf(fabsf(disc));
    const float invmu = (mode != 2) ? (1.0f / mu) : 0.0f;
    const float lp    = alpha + mu;   // overdamped eigenvalues
    const float lm    = alpha - mu;

    // Uniform amplitude-underflow cutoff: expf(alpha*dt) == +0.0f (past the
    // fp32 denormal floor) once alpha*dt < -104.  Lanes past the cutoff skip
    // expf AND sincosf entirely; the branch is wave-uniform except at the
    // single crossover wave of the whole grid.
    const float dt_cut = (alpha < 0.0f) ? (104.0f / (-alpha)) : 3.4e38f;

    const int i0     = (blockIdx.x * blockDim.x + threadIdx.x) * 4;
    const int stride = gridDim.x * blockDim.x * 4;

    for (int base = i0; base < n; base += stride) {
        // Stream-ahead prefetch of next grid-stride tile of t (global_prefetch_b8).
        if (base + stride + 3 < n)
            __builtin_prefetch(t + base + stride, 0, 1);

        float tt[4];
        const bool full = (base + 3 < n);
        if (full) {
            const float4 tv = *(const float4*)(t + base);
            tt[0] = tv.x; tt[1] = tv.y; tt[2] = tv.z; tt[3] = tv.w;
        } else {
            #pragma unroll
            for (int k = 0; k < 4; ++k)
                tt[k] = (base + k < n) ? t[base + k] : t0;
        }

        float o[8];
        #pragma unroll
        for (int k = 0; k < 4; ++k) {
            const float dt = tt[k] - t0;
            float u, v;
            if (mode == 0) {                       // underdamped (uniform branch)
                if (dt < dt_cut) {
                    const float e = expf(alpha * dt);
                    float s, c;
                    sincosf(mu * dt, &s, &c);
                    u = e * c;
                    v = e * (s * invmu);
                } else {                           // e^{alpha*dt} == 0 in fp32
                    u = 0.0f;
                    v = 0.0f;
                }
            } else if (mode == 1) {                // overdamped: via exponentials
                const float ep = expf(lp * dt);
                const float em = expf(lm * dt);
                u = 0.5f * (ep + em);
                v = 0.5f * (ep - em) * invmu;
            } else {                               // critically damped
                const float e = expf(alpha * dt);
                u = e;
                v = e * dt;
            }
            o[2 * k + 0] = u * w00 + v * w10;
            o[2 * k + 1] = u * w01 + v * w11;
        }

        if (full) {
            float4* dst = (float4*)(out + 2 * base);   // 16B-aligned (base % 4 == 0)
            dst[0] = make_float4(o[0], o[1], o[2], o[3]);
            dst[1] = make_float4(o[4], o[5], o[6], o[7]);
        } else {
            for (int k = 0; k < 4 && base + k < n; ++k) {
                out[2 * (base + k) + 0] = o[2 * k + 0];
                out[2 * (base + k) + 1] = o[2 * k + 1];
            }
        }
    }
}

extern "C" void kernel_launch(void* const* d_in, const int* in_sizes, int n_in,
                              void* d_out, int out_size, void* d_ws, size_t ws_size,
                              hipStream_t stream) {
    (void)n_in; (void)d_ws; (void)ws_size; (void)out_size;
    const float* t    = (const float*)d_in[0];   // [T] time grid
    const float* x0   = (const float*)d_in[1];   // [2]
    const float* om2  = (const float*)d_in[2];   // scalar
    const float* gam  = (const float*)d_in[3];   // scalar
    float* out        = (float*)d_out;           // [T,2] interleaved

    const int n = in_sizes[0];                   // T = 4,194,304

    const int threads = 256;
    // 4 elements/thread; cap blocks so each thread runs ~2 grid-stride
    // iterations (makes the stream-ahead prefetch meaningful).
    long long want = ((long long)n / 4 + threads - 1) / threads;
    int blocks = (int)(want < 1 ? 1 : (want > 2048 ? 2048 : want));

    magnus_rlc_traj_kernel<<<blocks, threads, 0, stream>>>(t, x0, om2, gam, out, n);
}